// GNNTimeModel_9655086482050
// MI455X (gfx1250) — compile-verified
//
#include <hip/hip_runtime.h>

typedef __attribute__((ext_vector_type(2))) float v2f;
typedef __attribute__((ext_vector_type(4))) float v4f;
typedef __attribute__((ext_vector_type(8))) float v8f;

#define B_DIM 64
#define V_DIM 64
#define T_DIM 256
#define F_DIM 64
#define LDSW  66   // padded row stride: bank = (2*row+col)%64, conflict-free

__launch_bounds__(512, 1)
__global__ void gnn_time_fused_kernel(const float* __restrict__ X,   // [B,V,T]
                                      const float* __restrict__ A,   // [T,V,V]
                                      const float* __restrict__ W,   // [F,1]
                                      float* __restrict__ out) {     // [B,T,V*F]
  __shared__ float Xs[V_DIM * LDSW];    // Xs[b][w]   = X[b, w, t]
  __shared__ float adjS[V_DIM * LDSW];  // adjS[v][w] = sigmoid-masked adjacency
  __shared__ float Hs[V_DIM * LDSW];    // Hs[b][v]   = H_agg tile for this t
  __shared__ v4f   Ws4[F_DIM / 4];

  const int t   = blockIdx.x;
  const int tid = threadIdx.x;

  // ---------------- Phase A: stage X slice + adjacency into LDS ------------
  if (tid < F_DIM) ((float*)Ws4)[tid] = W[tid];
#pragma unroll
  for (int i = 0; i < 8; ++i) {
    int idx = tid + i * 512;          // 0..4095
    int row = idx >> 6;               // b (for Xs) / v (for adjS)
    int col = idx & 63;               // w
    Xs[row * LDSW + col] = X[(row * V_DIM + col) * T_DIM + t];
    float a = A[(t * V_DIM + row) * V_DIM + col];
    float s = 1.0f / (1.0f + __expf(-a));
    adjS[row * LDSW + col] = (row == col) ? 1.0f : s;
  }
  __syncthreads();

  // ---------------- Phase B: 64x64x64 GEMM via v_wmma_f32_16x16x4_f32 -----
  // C[b][v] = sum_w Xs[b][w] * adjS[v][w]; 16 tiles of 16x16, one per wave.
  {
    const int wave  = tid >> 5;       // 0..15
    const int lane  = tid & 31;
    const int tileM = wave >> 2;      // b-tile 0..3
    const int tileN = wave & 3;       // v-tile 0..3
    const int l16   = lane & 15;
    const int khalf = (lane >> 4) * 2;  // lanes 0-15: K+{0,1}; 16-31: K+{2,3}
    const float* arow = &Xs[(tileM * 16 + l16) * LDSW];    // A: row = b
    const float* brow = &adjS[(tileN * 16 + l16) * LDSW];  // B^T row = v
    v8f acc = {};
#pragma unroll
    for (int k0 = 0; k0 < V_DIM; k0 += 4) {
      v2f a, b;
      a.x = arow[k0 + khalf];
      a.y = arow[k0 + khalf + 1];
      b.x = brow[k0 + khalf];
      b.y = brow[k0 + khalf + 1];
      acc = __builtin_amdgcn_wmma_f32_16x16x4_f32(false, a, false, b,
                                                  (short)0, acc, false, false);
    }
    // D layout: lane holds C[m][n], m = i + 8*(lane>>4), n = l16
    const int mbase = tileM * 16 + 8 * (lane >> 4);
    const int n     = tileN * 16 + l16;
#pragma unroll
    for (int i = 0; i < 8; ++i) Hs[(mbase + i) * LDSW + n] = acc[i];
  }
  __syncthreads();

  // ---------------- Phase C: streaming expand (the bandwidth phase) -------
  // out[b,t,v,f] = leaky(Hs[b][v] * W[f]); float4 non-temporal stores,
  // consecutive lanes -> consecutive 16B chunks (fully coalesced).
  const long long outT = (long long)t * (V_DIM * F_DIM);
  for (int chunk = tid; chunk < B_DIM * V_DIM * (F_DIM / 4); chunk += 512) {
    int f4 = chunk & 15;
    int v  = (chunk >> 4) & 63;
    int b  = chunk >> 10;
    float h = Hs[b * LDSW + v];
    v4f w4 = Ws4[f4];
    v4f r;
#pragma unroll
    for (int j = 0; j < 4; ++j) {
      float p = h * w4[j];
      r[j] = (p >= 0.0f) ? p : 0.01f * p;
    }
    long long o = (long long)b * (T_DIM * V_DIM * F_DIM) + outT + v * F_DIM + f4 * 4;
    __builtin_nontemporal_store(r, (v4f*)(out + o));
  }
}

extern "C" void kernel_launch(void* const* d_in, const int* in_sizes, int n_in,
                              void* d_out, int out_size, void* d_ws, size_t ws_size,
                              hipStream_t stream) {
  const float* X = (const float*)d_in[0];   // [64, 64, 256]
  const float* A = (const float*)d_in[1];   // [256, 64, 64]
  const float* W = (const float*)d_in[2];   // [64, 1]
  float* out = (float*)d_out;               // [64, 256, 64*64]
  (void)in_sizes; (void)n_in; (void)out_size; (void)d_ws; (void)ws_size;
  gnn_time_fused_kernel<<<T_DIM, 512, 0, stream>>>(X, A, W, out);
}